// Hbv4Dpl_3934190044236
// MI455X (gfx1250) — compile-verified
//
#include <hip/hip_runtime.h>
#include <math.h>

// ---------------------------------------------------------------------------
// HBV hydrological model, fused forward pass for MI455X (gfx1250).
//
// Problem shape:  T=2190 timesteps x G=10000 cells, 14 params/cell.
//   - serial per-cell recurrence  -> parallelism capped at 10000 threads
//   - ~340 MB total traffic       -> ~15us HBM floor, compute trivial
//   - per-cell FIR weights        -> no shared operand => WMMA inapplicable
//
// With ~313 wave32s chip-wide (~1 wave/SIMD) there is no TLP to hide memory
// latency, so we use CDNA5's async global->LDS engine (ASYNCcnt) to keep a
// deep double-buffered prefetch of the forcing data in flight while the VALU
// chain runs. 24 async b32 ops per TB=8-step buffer, <=48 outstanding
// (6-bit ASYNCcnt max 63), drained with s_wait_asynccnt 24.
//
// v2: hoisted loop-invariant reciprocals (1/fc, 1/(lp*fc)) and cfr*cfmax out
// of the time loop — removes two ~10-op f32 divide expansions from the
// serial per-step dependency chain, which is the binding resource here.
// ---------------------------------------------------------------------------

#define T_TOT    2190
#define G_TOT    10000
#define WARMUP_  365
#define KER      15
#define TB       8            // timesteps staged per LDS buffer
#define SLOT_DW  96           // 32 cells * 3 forcing vars, contiguous in memory
#define NWAVE    4
#define BLOCKSZ  (NWAVE * 32)

__global__ __launch_bounds__(BLOCKSZ, 1)
void hbv_fused_kernel(const float* __restrict__ x,     // (T, G, 3)
                      const float* __restrict__ prm,   // (G, 14) in [0,1]
                      float* __restrict__ out)         // (T-WARMUP, G)
{
    __shared__ float smem[NWAVE * 2 * TB * SLOT_DW];   // 24 KB / workgroup

    const int tid  = threadIdx.x;
    const int lane = tid & 31;
    const int wv   = tid >> 5;
    const int g    = blockIdx.x * BLOCKSZ + tid;
    const int g0   = blockIdx.x * BLOCKSZ + (tid & ~31);   // wave's first cell
    const int gp   = (g < G_TOT) ? g : (G_TOT - 1);        // clamp param reads

    const int      waveBaseDW  = wv * (2 * TB * SLOT_DW);
    const uint32_t ldsWaveByte = (uint32_t)(size_t)(&smem[waveBaseDW]);

    // ---- parameter scaling (PARAM_RANGES baked in) ----
    const float lo[14] = {1.0f, 50.0f, 0.05f, 0.01f, 0.001f, 0.2f, 0.0f,
                          0.0f, -2.5f, 0.5f, 0.0f, 0.0f, 0.0f, 0.0f};
    const float hi[14] = {6.0f, 1000.0f, 0.9f, 0.5f, 0.2f, 1.0f, 10.0f,
                          100.0f, 2.5f, 10.0f, 0.1f, 0.2f, 2.9f, 6.5f};
    float P[14];
#pragma unroll
    for (int i = 0; i < 14; ++i) {
        float u = prm[gp * 14 + i];
        P[i] = lo[i] + u * (hi[i] - lo[i]);
    }
    const float beta = P[0], fc = P[1], k0 = P[2], k1 = P[3], k2 = P[4];
    const float lp = P[5], pperc = P[6], uzl = P[7], ttp = P[8];
    const float cfmax = P[9], cfr = P[10], cwh = P[11];

    // loop-invariant hoists: keep f32 divides OFF the per-step serial chain
    const float rfc   = 1.0f / fc;            // for (sm/fc)^beta
    const float rlpfc = 1.0f / (lp * fc);     // for sm/(lp*fc)
    const float cfrm  = cfr * cfmax;          // refreeze coefficient

    // ---- gamma unit-hydrograph weights (once per cell) ----
    float w[KER];
    {
        float aa  = fmaxf(P[12], 0.0f) + 0.1f;
        float th  = fmaxf(P[13], 0.0f) + 0.5f;
        float rth = 1.0f / th;
        float lg  = lgammaf(aa);
        float lth = logf(th);
        float s   = 0.0f;
#pragma unroll
        for (int k = 0; k < KER; ++k) {
            float tv = (float)k + 0.5f;                       // 0.5 .. 14.5
            float lw = (aa - 1.0f) * logf(tv) - tv * rth - lg - aa * lth;
            w[k] = expf(lw);
            s += w[k];
        }
        float inv = 1.0f / s;
#pragma unroll
        for (int k = 0; k < KER; ++k) w[k] *= inv;
    }

    // ---- async stage: TB timesteps of 96 contiguous dwords into LDS ----
    // All 32 lanes always issue (addresses clamped in-bounds) so ASYNCcnt
    // bookkeeping stays exact even for the partial final wave.
    auto fill = [&](int buf, int bb) {
        // prior ds_loads from this buffer's last use must land before the
        // async engine overwrites the region
        asm volatile("s_wait_dscnt 0" ::: "memory");
#pragma unroll
        for (int i = 0; i < TB; ++i) {
            int tt = bb * TB + i;
            if (tt > T_TOT - 1) tt = T_TOT - 1;               // clamp tail
            long long dwBase = ((long long)tt * G_TOT + g0) * 3;
            uint32_t laddr0 = ldsWaveByte +
                              (uint32_t)((buf * TB + i) * SLOT_DW + lane) * 4u;
#pragma unroll
            for (int c = 0; c < 3; ++c) {
                long long idx = dwBase + (long long)(c * 32 + lane);
                const long long mx = (long long)T_TOT * G_TOT * 3 - 1;
                if (idx > mx) idx = mx;                       // stay in bounds
                uint32_t goff = (uint32_t)idx << 2;           // byte offset < 4GB
                uint32_t la   = laddr0 + (uint32_t)(c * 128);
                // GVS mode: mem = SADDR(64b base) + VADDR(32b byte offset)
                asm volatile("global_load_async_to_lds_b32 %0, %1, %2"
                             :: "v"(la), "v"(goff), "s"(x)
                             : "memory");
            }
        }
    };

    // ---- state ----
    float snowpack = 0.001f, meltwater = 0.001f, sm = 0.001f;
    float suz = 0.001f, slz = 0.001f;
    float fir[KER - 1];                 // transposed-form FIR delay line
#pragma unroll
    for (int k = 0; k < KER - 1; ++k) fir[k] = 0.0f;

    fill(0, 0);                         // prologue: buffer 0 <- block 0

    const int NB = (T_TOT + TB - 1) / TB;
    for (int b = 0; b < NB; ++b) {
        fill((b + 1) & 1, b + 1);       // prefetch next block (clamped at end)
        // 24 newest ops belong to the prefetch; once <=24 remain, buffer b
        // is fully landed (async loads complete in order).
        asm volatile("s_wait_asynccnt 24" ::: "memory");

        const int cur = b & 1;
        const float* bufp = &smem[waveBaseDW + cur * TB * SLOT_DW + lane * 3];

        for (int i = 0; i < TB; ++i) {
            int t = b * TB + i;
            if (t >= T_TOT) break;

            float precip = bufp[i * SLOT_DW + 0];
            float potent = bufp[i * SLOT_DW + 1];
            float tempre = bufp[i * SLOT_DW + 2];

            // ---- HBV step (all clamps preserve reference semantics) ----
            float is_rain = (tempre >= ttp) ? 1.0f : 0.0f;
            float rain = precip * is_rain;
            float snow = precip - rain;
            snowpack += snow;
            float melt = fminf(fmaxf(cfmax * (tempre - ttp), 0.0f), snowpack);
            meltwater += melt;
            snowpack  -= melt;
            float refreezing =
                fminf(fmaxf(cfrm * (ttp - tempre), 0.0f), meltwater);
            snowpack  += refreezing;
            meltwater -= refreezing;
            float to_soil = fmaxf(meltwater - cwh * snowpack, 0.0f);
            meltwater -= to_soil;
            float soil_wetness =
                fminf(fmaxf(__powf(sm * rfc, beta), 0.0f), 1.0f);
            float recharge = (rain + to_soil) * soil_wetness;
            sm += rain + to_soil - recharge;
            float excess = fmaxf(sm - fc, 0.0f);
            sm -= excess;
            float evap_factor = fminf(fmaxf(sm * rlpfc, 0.0f), 1.0f);
            float et_act = fminf(sm, potent * evap_factor);
            sm = fmaxf(sm - et_act, 1e-5f);
            suz += recharge + excess;
            float perc = fminf(suz, pperc);
            suz -= perc;
            float q0v = k0 * fmaxf(suz - uzl, 0.0f);
            suz -= q0v;
            float q1v = k1 * suz;
            suz -= q1v;
            slz += perc;
            float q2v = k2 * slz;
            slz -= q2v;

            // ---- UH convolution (transposed-form FIR), post-warmup only ----
            if (t >= WARMUP_) {
                float q = q0v + q1v + q2v;
                float y = fmaf(w[0], q, fir[0]);
#pragma unroll
                for (int k = 0; k < KER - 2; ++k)
                    fir[k] = fmaf(w[k + 1], q, fir[k + 1]);
                fir[KER - 2] = w[KER - 1] * q;
                if (g < G_TOT)
                    out[(size_t)(t - WARMUP_) * G_TOT + g] = y;
            }
        }
    }
}

extern "C" void kernel_launch(void* const* d_in, const int* in_sizes, int n_in,
                              void* d_out, int out_size, void* d_ws, size_t ws_size,
                              hipStream_t stream) {
    (void)in_sizes; (void)n_in; (void)out_size; (void)d_ws; (void)ws_size;
    const float* x   = (const float*)d_in[0];   // (2190, 10000, 3) f32
    const float* prm = (const float*)d_in[1];   // (10000, 14) f32
    float* out = (float*)d_out;                 // (1825, 10000, 1) f32

    dim3 grid((G_TOT + BLOCKSZ - 1) / BLOCKSZ); // 79 blocks
    dim3 block(BLOCKSZ);                        // 128 threads = 4 wave32s
    hipLaunchKernelGGL(hbv_fused_kernel, grid, block, 0, stream, x, prm, out);
}